// PolymerJEPAv2_6433861010010
// MI455X (gfx1250) — compile-verified
//
#include <hip/hip_runtime.h>
#include <hip/hip_bf16.h>
#include <stdint.h>

// ---------------- problem constants (fixed by the reference) ----------------
static constexpr int   N_   = 200000;
static constexpr int   E_   = 800000;
static constexpr int   NS_  = 400000;
static constexpr long  ES_  = 1600000;
static constexpr int   B_   = 512;
static constexpr int   PPG_ = 8;
static constexpr int   PT_  = 4;
static constexpr int   NF_  = 64;
static constexpr int   NE_  = 16;
static constexpr int   H_   = 128;
static constexpr int   RW_  = 16;
static constexpr int   PRW_ = 16;
static constexpr int   L_   = 4;
static constexpr int   P_   = B_ * PPG_;      // 4096 patches
static constexpr int   BT_  = B_ * PT_;       // 2048 target rows

typedef __attribute__((ext_vector_type(2))) float v2f;
typedef __attribute__((ext_vector_type(8))) float v8f;

// ---------------------------------------------------------------------------
// Pair-interleave a weight matrix W[K,128] so a WMMA B-fragment
// (W[kk][col], W[kk+1][col]) is one contiguous 8-byte LDS load:
//   Wdst[(k>>1)*256 + col*2 + (k&1)] = Wsrc[k*128 + col]
// ---------------------------------------------------------------------------
__global__ void interleave_w(const float* __restrict__ Wsrc, float* __restrict__ Wdst, int K) {
  int i = blockIdx.x * blockDim.x + threadIdx.x;
  if (i >= K * 128) return;
  int k = i >> 7, col = i & 127;
  Wdst[(k >> 1) * 256 + col * 2 + (k & 1)] = Wsrc[i];
}

// ---------------------------------------------------------------------------
// Generic fused GEMM: OUT[M,128] = act( A1[g(r),0:K1] @ W1 + A2[g(r),0:K2] @ W2 + bias )
// W1/W2 are PRE-INTERLEAVED [K,128] blocks (see interleave_w). Weights are
// staged into LDS with gfx1250 async-to-LDS DMA; one wave computes a 16x128
// output strip with V_WMMA_F32_16X16X4_F32 (full fp32; model is memory-bound,
// so fp32 matrix ops cost nothing at the roofline and keep exact precision).
// B fragments are single ds_load_b64 ops thanks to the interleaved layout.
// ---------------------------------------------------------------------------
template <int K1, int K2, bool RELU>
__global__ __launch_bounds__(256) void gemm_h128(
    const float* __restrict__ A1, const float* __restrict__ A2,
    const float* __restrict__ W1, const float* __restrict__ W2,
    const float* __restrict__ bias, float* __restrict__ out,
    int M, const int* __restrict__ gather)
{
  constexpr int KMAX = (K1 > K2) ? K1 : K2;
  __shared__ float Wlds[KMAX * 128];

  const int lane  = threadIdx.x & 31;
  const int wave  = threadIdx.x >> 5;
  const int row0  = (blockIdx.x * 8 + wave) * 16;
  const bool active = (row0 < M);              // wave-uniform -> EXEC all-ones at WMMA
  const int mrow  = lane & 15;                 // A row / C col within tile
  const int khalf = (lane >> 4) << 1;          // 0 or 2 (K sub-pair per half-wave)

  long arow = 0;
  if (active) {
    const int r = row0 + mrow;
    arow = gather ? (long)gather[r] : (long)r;
  }

  v8f acc[8] = {};

#pragma unroll
  for (int s = 0; s < 2; ++s) {
    const int K = (s == 0) ? K1 : K2;
    if (K == 0) break;
    const float* Wg = (s == 0) ? W1 : W2;
    const float* Ag = (s == 0) ? A1 : A2;
    if (s == 1) __syncthreads();               // protect LDS reuse across stages

    // --- stage (interleaved) weights into LDS with async DMA (gfx1250) ---
    {
      const int n16 = (K * 128) / 4;           // float4 chunks
      for (int i = threadIdx.x; i < n16; i += 256) {
        uint32_t la = (uint32_t)(uint64_t)(const void*)&Wlds[i * 4];
        const float* gp = Wg + i * 4;
        asm volatile("global_load_async_to_lds_b128 %0, %1, off"
                     :: "v"(la), "v"(gp) : "memory");
      }
      asm volatile("s_wait_asynccnt 0" ::: "memory");
    }
    __syncthreads();

    if (active) {
      const float* ap = Ag + arow * K;
      for (int k = 0; k < K; k += 4) {
        v2f a = *(const v2f*)(ap + k + khalf);             // 8B global load
        const float* brow = &Wlds[((k + khalf) >> 1) * 256];
#pragma unroll
        for (int nt = 0; nt < 8; ++nt) {
          const int col = nt * 16 + mrow;
          v2f b = *(const v2f*)(brow + col * 2);           // one ds_load_b64
          acc[nt] = __builtin_amdgcn_wmma_f32_16x16x4_f32(
              false, a, false, b, (short)0, acc[nt], false, false);
        }
      }
    }
  }

  if (active) {
#pragma unroll
    for (int nt = 0; nt < 8; ++nt) {
      const int col = nt * 16 + mrow;
      const float bv = bias[col];
#pragma unroll
      for (int j = 0; j < 8; ++j) {
        const long m = row0 + j + 8 * (lane >> 4);
        float v = acc[nt][j] + bv;
        if (RELU) v = fmaxf(v, 0.0f);
        out[m * 128 + col] = v;
      }
    }
  }
}

// ---------------------------------------------------------------------------
// Elementwise / scatter helpers
// ---------------------------------------------------------------------------
__global__ void zero_f(float* __restrict__ p, long n) {
  long i = (long)blockIdx.x * blockDim.x + threadIdx.x;
  if (i < n) p[i] = 0.0f;
}

__global__ void vec_add2(const float* __restrict__ a, const float* __restrict__ b,
                         float* __restrict__ o, int n) {
  int i = blockIdx.x * blockDim.x + threadIdx.x;
  if (i < n) o[i] = a[i] + b[i];
}

// dst[r,:] = src[idx[r],:]   (H=128, float4 per thread)
__global__ void gather_rows(const float* __restrict__ src, const int* __restrict__ idx,
                            float* __restrict__ dst, long nrows) {
  long t = (long)blockIdx.x * blockDim.x + threadIdx.x;
  if (t >= nrows * 32) return;
  long row = t >> 5;
  int c4 = (int)(t & 31) * 4;
  long sr = idx[row];
  *(float4*)(dst + row * 128 + c4) = *(const float4*)(src + sr * 128 + c4);
}

// agg[dst[e],:] += (h[src[e],:] + e_feat[e,:]) * w[e]
__global__ void msg_scatter(const float* __restrict__ h, const float* __restrict__ ef,
                            const int* __restrict__ src, const int* __restrict__ dst,
                            const float* __restrict__ ew, const int* __restrict__ ewmap,
                            float* __restrict__ agg, long ne) {
  long t = (long)blockIdx.x * blockDim.x + threadIdx.x;
  if (t >= ne * 32) return;
  long e = t >> 5;
  int c4 = (int)(t & 31) * 4;
  int s = src[e], d = dst[e];
  float w = ewmap ? ew[ewmap[e]] : ew[e];
  float4 hv = *(const float4*)(h + (long)s * 128 + c4);
  float4 ev = *(const float4*)(ef + e * 128 + c4);
  float* ap = agg + (long)d * 128 + c4;
  atomicAdd(ap + 0, (hv.x + ev.x) * w);
  atomicAdd(ap + 1, (hv.y + ev.y) * w);
  atomicAdd(ap + 2, (hv.z + ev.z) * w);
  atomicAdd(ap + 3, (hv.w + ev.w) * w);
}

// segment sum + count:  sums[seg[r],:] += v[g(r),:] ; cnt[seg[r]] += 1
__global__ void seg_sum(const float* __restrict__ v, const int* __restrict__ gather,
                        const int* __restrict__ seg, float* __restrict__ sums,
                        float* __restrict__ cnt, long nrows) {
  long t = (long)blockIdx.x * blockDim.x + threadIdx.x;
  if (t >= nrows * 32) return;
  long row = t >> 5;
  int c4 = (int)(t & 31) * 4;
  int b = seg[row];
  long sr = gather ? (long)gather[row] : row;
  float4 x = *(const float4*)(v + sr * 128 + c4);
  float* p = sums + (long)b * 128 + c4;
  atomicAdd(p + 0, x.x);
  atomicAdd(p + 1, x.y);
  atomicAdd(p + 2, x.z);
  atomicAdd(p + 3, x.w);
  if ((t & 31) == 0) atomicAdd(cnt + b, 1.0f);
}

__global__ void seg_div(float* __restrict__ sums, const float* __restrict__ cnt, int total) {
  int i = blockIdx.x * blockDim.x + threadIdx.x;
  if (i < total) sums[i] /= fmaxf(cnt[i >> 7], 1.0f);
}

__global__ void make_tidx(const int* __restrict__ tgt_idxs, int* __restrict__ tidx) {
  int i = blockIdx.x * blockDim.x + threadIdx.x;
  if (i >= BT_) return;
  int b = i >> 2;                       // PT = 4
  tidx[i] = tgt_idxs[i] + b * PPG_;
}

// emb_tgt rows + cond rows (cond = emb_ctx[b] + enc_pes[row])
__global__ void assemble(const float* __restrict__ emb_sub, const float* __restrict__ sub_full,
                         const int* __restrict__ ctx_idx, const int* __restrict__ tidx,
                         const float* __restrict__ enc, float* __restrict__ out_embtgt,
                         float* __restrict__ cond) {
  int i = blockIdx.x * blockDim.x + threadIdx.x;
  if (i >= BT_ * 128) return;
  int row = i >> 7, col = i & 127;
  int b = row >> 2;
  int ti = tidx[row];
  out_embtgt[i] = sub_full[(long)ti * 128 + col];
  cond[i] = emb_sub[(long)(ctx_idx[b] + b * PPG_) * 128 + col] + enc[i];
}

// per-column mean / rsqrt(var+eps) over R rows (biased var, matches jnp)
__global__ __launch_bounds__(256) void col_stats(const float* __restrict__ z, int R,
                                                 float* __restrict__ mean,
                                                 float* __restrict__ rstd) {
  __shared__ float ssum[256], ssq[256];
  int col = blockIdx.x;
  float s = 0.0f, q = 0.0f;
  for (int r = threadIdx.x; r < R; r += 256) {
    float v = z[(long)r * 128 + col];
    s += v; q += v * v;
  }
  ssum[threadIdx.x] = s; ssq[threadIdx.x] = q;
  __syncthreads();
  for (int o = 128; o > 0; o >>= 1) {
    if (threadIdx.x < o) { ssum[threadIdx.x] += ssum[threadIdx.x + o];
                           ssq[threadIdx.x]  += ssq[threadIdx.x + o]; }
    __syncthreads();
  }
  if (threadIdx.x == 0) {
    float m = ssum[0] / (float)R;
    float var = ssq[0] / (float)R - m * m;
    mean[col] = m;
    rstd[col] = rsqrtf(var + 1e-5f);
  }
}

__global__ void bn_relu(const float* __restrict__ z, const float* __restrict__ mean,
                        const float* __restrict__ rstd, const float* __restrict__ g,
                        const float* __restrict__ b, float* __restrict__ o, long n) {
  long i = (long)blockIdx.x * blockDim.x + threadIdx.x;
  if (i >= n) return;
  int c = (int)(i & 127);
  o[i] = fmaxf((z[i] - mean[c]) * rstd[c] * g[c] + b[c], 0.0f);
}

// ---------------------------------------------------------------------------
static inline int cdivl(long a, long b) { return (int)((a + b - 1) / b); }

extern "C" void kernel_launch(void* const* d_in, const int* in_sizes, int n_in,
                              void* d_out, int out_size, void* d_ws, size_t ws_size,
                              hipStream_t stream) {
  (void)in_sizes; (void)n_in; (void)out_size; (void)ws_size;

  const float* x        = (const float*)d_in[0];
  const float* rw       = (const float*)d_in[1];
  /* d_in[2] node_weight: unused by reference */
  const int*   edge_index = (const int*)d_in[3];
  const float* edge_attr  = (const float*)d_in[4];
  const float* edge_weight= (const float*)d_in[5];
  const float* patch_pe   = (const float*)d_in[6];
  const int*   nmap    = (const int*)d_in[7];
  const int*   csub    = (const int*)d_in[8];
  const int*   emap    = (const int*)d_in[9];
  const int*   sbatch  = (const int*)d_in[10];
  const int*   ctx_idx = (const int*)d_in[11];
  const int*   tgtidx  = (const int*)d_in[12];
  /* d_in[13] patches_per_graph: fixed = 8 */
  const float* W_in = (const float*)d_in[14]; const float* b_in = (const float*)d_in[15];
  const float* W_rw = (const float*)d_in[16]; const float* b_rw = (const float*)d_in[17];
  const float* W_prw= (const float*)d_in[18]; const float* b_prw= (const float*)d_in[19];
  const float* ctx_We=(const float*)d_in[20]; const float* ctx_be=(const float*)d_in[21];
  const float* ctx_Wl=(const float*)d_in[22]; const float* ctx_bl=(const float*)d_in[23];
  const float* tgt_We=(const float*)d_in[24]; const float* tgt_be=(const float*)d_in[25];
  const float* tgt_Wl=(const float*)d_in[26]; const float* tgt_bl=(const float*)d_in[27];
  const float* Wp1=(const float*)d_in[28]; const float* bp1=(const float*)d_in[29];
  const float* g1 =(const float*)d_in[30]; const float* be1=(const float*)d_in[31];
  const float* Wp2=(const float*)d_in[32]; const float* bp2=(const float*)d_in[33];
  const float* g2 =(const float*)d_in[34]; const float* be2=(const float*)d_in[35];
  const float* Wp3=(const float*)d_in[36]; const float* bp3=(const float*)d_in[37];

  float* out = (float*)d_out;                // [2, B, PT, H]; emb_tgt then pred
  float* out_embtgt = out;
  float* out_pred   = out + (long)BT_ * H_;

  // ------------------- workspace layout -------------------
  float* w = (float*)d_ws;
  size_t off = 0;
  float* x0      = w + off; off += (size_t)N_ * H_;     // node encode (kept)
  float* e_buf   = w + off; off += (size_t)ES_ * H_;    // ctx edge feats; later tgt edge feats
  float* ha      = w + off; off += (size_t)NS_ * H_;    // ping  (ctx), reused by tgt
  float* hb      = w + off; off += (size_t)NS_ * H_;    // pong
  float* agg     = w + off; off += (size_t)NS_ * H_;    // scatter accumulator
  float* emb_sub = w + off; off += (size_t)P_ * H_;
  float* sub_full= w + off; off += (size_t)P_ * H_;
  float* cnt1    = w + off; off += P_;
  float* cnt2    = w + off; off += P_;
  float* enc     = w + off; off += (size_t)BT_ * H_;
  float* cond    = w + off; off += (size_t)BT_ * H_;
  float* z1      = w + off; off += (size_t)BT_ * H_;
  float* p1      = w + off; off += (size_t)BT_ * H_;
  float* z2      = w + off; off += (size_t)BT_ * H_;
  float* p2      = w + off; off += (size_t)BT_ * H_;
  float* wr1     = w + off; off += (size_t)H_ * H_;     // interleaved weight slot A
  float* wr2     = w + off; off += (size_t)H_ * H_;     // interleaved weight slot B
  float* mean1   = w + off; off += H_;
  float* rstd1   = w + off; off += H_;
  float* mean2   = w + off; off += H_;
  float* rstd2   = w + off; off += H_;
  float* bias2   = w + off; off += H_;
  int*   tidx    = (int*)(w + off); off += BT_;

  const int TB = 256;
  auto ilv = [&](const float* Wsrc, float* Wdst, int K) {
    interleave_w<<<cdivl((long)K * 128, TB), TB, 0, stream>>>(Wsrc, Wdst, K);
  };

  // --- shared node embedding: x0 = x@W_in + rw@W_rw + (b_in + b_rw) ---
  vec_add2<<<1, 128, 0, stream>>>(b_in, b_rw, bias2, H_);
  ilv(W_in, wr1, NF_);
  ilv(W_rw, wr2, RW_);
  gemm_h128<NF_, RW_, false><<<cdivl(N_, 128), TB, 0, stream>>>(
      x, rw, wr1, wr2, bias2, x0, N_, nullptr);

  // ================= context branch (batched subgraphs) =================
  gather_rows<<<cdivl((long)NS_ * 32, TB), TB, 0, stream>>>(x0, nmap, ha, NS_);
  ilv(ctx_We, wr1, NE_);
  gemm_h128<NE_, 0, true><<<cdivl(ES_, 128), TB, 0, stream>>>(
      edge_attr, nullptr, wr1, nullptr, ctx_be, e_buf, (int)ES_, emap);

  const int* cs_src = csub;
  const int* cs_dst = csub + ES_;
  float* hc = ha; float* hn = hb;
  for (int l = 0; l < L_; ++l) {
    zero_f<<<cdivl((long)NS_ * H_, TB), TB, 0, stream>>>(agg, (long)NS_ * H_);
    msg_scatter<<<cdivl(ES_ * 32, TB), TB, 0, stream>>>(
        hc, e_buf, cs_src, cs_dst, edge_weight, emap, agg, ES_);
    const float* Wl = ctx_Wl + (size_t)l * 2 * H_ * H_;
    ilv(Wl, wr1, H_);
    ilv(Wl + (size_t)H_ * H_, wr2, H_);
    gemm_h128<H_, H_, true><<<cdivl(NS_, 128), TB, 0, stream>>>(
        hc, agg, wr1, wr2, ctx_bl + (size_t)l * H_, hn, NS_, nullptr);
    float* t = hc; hc = hn; hn = t;
  }
  // hc holds final context node states
  zero_f<<<cdivl((long)P_ * H_, TB), TB, 0, stream>>>(emb_sub, (long)P_ * H_);
  zero_f<<<cdivl((long)P_, TB), TB, 0, stream>>>(cnt1, P_);
  seg_sum<<<cdivl((long)NS_ * 32, TB), TB, 0, stream>>>(hc, nullptr, sbatch, emb_sub, cnt1, NS_);
  seg_div<<<cdivl((long)P_ * H_, TB), TB, 0, stream>>>(emb_sub, cnt1, P_ * H_);

  // ================= target branch (full graph) =================
  ilv(tgt_We, wr1, NE_);
  gemm_h128<NE_, 0, true><<<cdivl(E_, 128), TB, 0, stream>>>(
      edge_attr, nullptr, wr1, nullptr, tgt_be, e_buf, E_, nullptr);

  const int* ei_src = edge_index;
  const int* ei_dst = edge_index + E_;
  const float* ht = x0; float* htn = ha;    // reuse ctx ping/pong buffers
  for (int l = 0; l < L_; ++l) {
    zero_f<<<cdivl((long)N_ * H_, TB), TB, 0, stream>>>(agg, (long)N_ * H_);
    msg_scatter<<<cdivl((long)E_ * 32, TB), TB, 0, stream>>>(
        ht, e_buf, ei_src, ei_dst, edge_weight, nullptr, agg, E_);
    const float* Wl = tgt_Wl + (size_t)l * 2 * H_ * H_;
    ilv(Wl, wr1, H_);
    ilv(Wl + (size_t)H_ * H_, wr2, H_);
    gemm_h128<H_, H_, true><<<cdivl(N_, 128), TB, 0, stream>>>(
        ht, agg, wr1, wr2, tgt_bl + (size_t)l * H_, htn, N_, nullptr);
    ht = htn;
    htn = (htn == ha) ? hb : ha;
  }
  // ht holds final full-graph node states
  zero_f<<<cdivl((long)P_ * H_, TB), TB, 0, stream>>>(sub_full, (long)P_ * H_);
  zero_f<<<cdivl((long)P_, TB), TB, 0, stream>>>(cnt2, P_);
  seg_sum<<<cdivl((long)NS_ * 32, TB), TB, 0, stream>>>(ht, nmap, sbatch, sub_full, cnt2, NS_);
  seg_div<<<cdivl((long)P_ * H_, TB), TB, 0, stream>>>(sub_full, cnt2, P_ * H_);

  // ================= head =================
  make_tidx<<<cdivl(BT_, TB), TB, 0, stream>>>(tgtidx, tidx);
  ilv(W_prw, wr1, PRW_);
  gemm_h128<PRW_, 0, false><<<cdivl(BT_, 128), TB, 0, stream>>>(
      patch_pe, nullptr, wr1, nullptr, b_prw, enc, BT_, tidx);
  assemble<<<cdivl((long)BT_ * H_, TB), TB, 0, stream>>>(
      emb_sub, sub_full, ctx_idx, tidx, enc, out_embtgt, cond);

  ilv(Wp1, wr1, H_);
  gemm_h128<H_, 0, false><<<cdivl(BT_, 128), TB, 0, stream>>>(
      cond, nullptr, wr1, nullptr, bp1, z1, BT_, nullptr);
  col_stats<<<H_, 256, 0, stream>>>(z1, BT_, mean1, rstd1);
  bn_relu<<<cdivl((long)BT_ * H_, TB), TB, 0, stream>>>(
      z1, mean1, rstd1, g1, be1, p1, (long)BT_ * H_);

  ilv(Wp2, wr1, H_);
  gemm_h128<H_, 0, false><<<cdivl(BT_, 128), TB, 0, stream>>>(
      p1, nullptr, wr1, nullptr, bp2, z2, BT_, nullptr);
  col_stats<<<H_, 256, 0, stream>>>(z2, BT_, mean2, rstd2);
  bn_relu<<<cdivl((long)BT_ * H_, TB), TB, 0, stream>>>(
      z2, mean2, rstd2, g2, be2, p2, (long)BT_ * H_);

  ilv(Wp3, wr1, H_);
  gemm_h128<H_, 0, false><<<cdivl(BT_, 128), TB, 0, stream>>>(
      p2, nullptr, wr1, nullptr, bp3, out_pred, BT_, nullptr);
}